// GATClassifier_51213190037805
// MI455X (gfx1250) — compile-verified
//
#include <hip/hip_runtime.h>

// ---------------------------------------------------------------------------
// GAT classifier for MI455X (gfx1250, wave32).
//   * GEMMs: V_WMMA_F32_16X16X4_F32 (exact f32 CDNA5 matrix path)
//   * A-tiles streamed to LDS by the Tensor Data Mover (tensor_load_to_lds,
//     TENSORcnt, double-buffered), with D# padding giving a conflict-free
//     LDS pitch of 68 floats.
//   * Edge softmax/aggregation via atomic passes (HBM-bandwidth bound).
// ---------------------------------------------------------------------------

#define N_NODES  20000
#define N_EDGES  320000
#define N_GRAPHS 64
#define DHEAD    128          // per-head feature dim (all 3 layers)

typedef __attribute__((ext_vector_type(2))) float v2f;
typedef __attribute__((ext_vector_type(8))) float v8f;

// -------- TDM availability (this toolchain: 6-arg builtin, clang-23 form) ---
#ifndef USE_TDM
#if defined(__has_builtin)
#if __has_builtin(__builtin_amdgcn_tensor_load_to_lds)
#define USE_TDM 1
#endif
#endif
#endif
#ifndef USE_TDM
#define USE_TDM 0
#endif

#if USE_TDM
typedef __attribute__((ext_vector_type(4))) unsigned tdm_u32x4;
typedef __attribute__((ext_vector_type(8))) int      tdm_i32x8;
typedef __attribute__((ext_vector_type(4))) int      tdm_i32x4;

// 2D tile load: 16 rows x KC cols of f32, row stride = row_stride_elems,
// LDS destination padded by D# pad fields to pitch 68 DWORDs per row.
__device__ __forceinline__ void tdm_load_16xKC(const float* gsrc, unsigned lds_off,
                                               int cols, int row_stride_elems) {
  unsigned long long ga = (unsigned long long)(uintptr_t)gsrc;
  tdm_u32x4 g0;
  g0[0] = 1u;                                    // count=1, user load D#
  g0[1] = lds_off;                               // LDS byte address
  g0[2] = (unsigned)(ga & 0xffffffffu);          // global_addr[31:0]
  g0[3] = (unsigned)((ga >> 32) & 0x01ffffffu)   // global_addr[56:32]
        | (2u << 30);                            // type = 2 ("image")

  const unsigned long long td0 = (unsigned)cols;            // tensor_dim0
  const unsigned long long td1 = 16ull;                     // tensor_dim1
  const unsigned long long st0 = (unsigned long long)row_stride_elems;
  // group1 word0: data_size=2 (4B) @16, pad_enable @20,
  //               pad_interval=5 (64 DW) @22, pad_amount=3 (4 DW) @25
  unsigned long long q0 = (2ull << 16) | (1ull << 20) | (5ull << 22) | (3ull << 25)
                        | (td0 << 48);                       // tensor_dim0[15:0]
  unsigned long long q1 = (td0 >> 16)                        // tensor_dim0[31:16]
                        | (td1 << 16)                        // tensor_dim1
                        | ((unsigned long long)(unsigned)cols << 48); // tile_dim0
  unsigned long long q2 = 16ull                              // tile_dim1
                        | ((st0 & 0xffffffffull) << 32);     // stride0[31:0]
  unsigned long long q3 = (st0 >> 32) & 0xffffull;           // stride0[47:32]
  tdm_i32x8 g1;
  g1[0]=(int)q0; g1[1]=(int)(q0>>32); g1[2]=(int)q1; g1[3]=(int)(q1>>32);
  g1[4]=(int)q2; g1[5]=(int)(q2>>32); g1[6]=(int)q3; g1[7]=(int)(q3>>32);
  tdm_i32x4 z4; z4[0]=0; z4[1]=0; z4[2]=0; z4[3]=0;          // 2D: groups 2/3 unused
  tdm_i32x8 z8; z8[0]=0; z8[1]=0; z8[2]=0; z8[3]=0; z8[4]=0; z8[5]=0; z8[6]=0; z8[7]=0;
  __builtin_amdgcn_tensor_load_to_lds(g0, g1, z4, z4, z8, 0);
}
#endif

// ---- order-preserving float<->uint mapping for atomic float max ------------
__device__ __forceinline__ unsigned f2ord(float f) {
  unsigned u = __float_as_uint(f);
  return (u & 0x80000000u) ? ~u : (u | 0x80000000u);
}
__device__ __forceinline__ float ord2f(unsigned u) {
  u = (u & 0x80000000u) ? (u & 0x7FFFFFFFu) : ~u;
  return __uint_as_float(u);
}
#define NEG_INF_ORD 0x007FFFFFu   // f2ord(-inf)

__device__ __forceinline__ float leaky02(float v) { return v > 0.0f ? v : 0.2f * v; }

__device__ __forceinline__ void edge_sd(const long long* __restrict__ ei, int e,
                                        int& s, int& d) {
  if (e < N_EDGES) { s = (int)ei[e]; d = (int)ei[N_EDGES + e]; }
  else             { s = e - N_EDGES; d = s; }            // self loops
}

// ---------------------------------------------------------------------------
// GEMM: H[N, Fout] = X[N, K] @ W[K, Fout]
//   block: 256 threads (8 waves), one 16-row tile, 8 x 16-col tiles
//   K streamed in 64-chunks, double-buffered LDS fed by TDM.
// Requires: N % 16 == 0, K % 64 == 0, Fout % 128 == 0 (true for all layers)
// ---------------------------------------------------------------------------
#define KC        64
#define LDS_PITCH 68    // 64 + 4 pad DWORDs -> banks 4r+c..+3 cover all 64 banks

__global__ __launch_bounds__(256) void gemm_wmma_f32(
    const float* __restrict__ X, const float* __restrict__ W,
    float* __restrict__ H, int K, int Fout) {
  __shared__ float As[2][16][LDS_PITCH];
  const int tid  = threadIdx.x;
  const int lane = tid & 31;
  const int wave = tid >> 5;
  const int half = lane >> 4;               // 0: lanes 0-15, 1: lanes 16-31
  const int r    = lane & 15;
  const int m0   = blockIdx.x * 16;
  const int n0   = blockIdx.y * 128 + wave * 16;
  const int nchunk = K / KC;

  auto load_chunk = [&](int c, int b) {
#if USE_TDM
    if (wave == 0)
      tdm_load_16xKC(X + (size_t)m0 * K + c * KC,
                     (unsigned)(uintptr_t)&As[b][0][0], KC, K);
#else
    #pragma unroll
    for (int i = tid; i < 16 * KC; i += 256) {
      const int rr = i >> 6, cc = i & (KC - 1);
      As[b][rr][cc] = X[(size_t)(m0 + rr) * K + c * KC + cc];
    }
#endif
  };

  v8f acc = {};
  load_chunk(0, 0);
  for (int c = 0; c < nchunk; ++c) {
    const int b = c & 1;
    if (c + 1 < nchunk) {
      load_chunk(c + 1, b ^ 1);
#if USE_TDM
      if (wave == 0) __builtin_amdgcn_s_wait_tensorcnt(1);  // chunk c landed
#endif
    } else {
#if USE_TDM
      if (wave == 0) __builtin_amdgcn_s_wait_tensorcnt(0);
#endif
    }
    __syncthreads();

    const int k0 = c * KC;
    #pragma unroll
    for (int kk = 0; kk < KC; kk += 4) {
      // A frag (16x4 f32): lanes 0-15 -> K=kk..kk+1, lanes 16-31 -> K=kk+2..kk+3
      v2f a, bf;
      a.x = As[b][r][kk + 2 * half + 0];
      a.y = As[b][r][kk + 2 * half + 1];
      // B frag (4x16 f32): K split across lane halves, N striped on lanes
      const float* Wp = W + (size_t)(k0 + kk + 2 * half) * Fout + (n0 + r);
      bf.x = Wp[0];
      bf.y = Wp[Fout];
      acc = __builtin_amdgcn_wmma_f32_16x16x4_f32(
          false, a, false, bf, (short)0, acc, false, false);
    }
    __syncthreads();
  }
  // C/D layout: VGPR v -> row m0+v (lanes 0-15) or m0+v+8 (lanes 16-31)
  #pragma unroll
  for (int v = 0; v < 8; ++v)
    H[(size_t)(m0 + v + 8 * half) * Fout + (n0 + r)] = acc[v];
}

// ---------------------------------------------------------------------------
__global__ void fill_f32(float* __restrict__ p, float v, size_t n) {
  size_t i = (size_t)blockIdx.x * blockDim.x + threadIdx.x;
  if (i < n) p[i] = v;
}
__global__ void fill_u32(unsigned* __restrict__ p, unsigned v, size_t n) {
  size_t i = (size_t)blockIdx.x * blockDim.x + threadIdx.x;
  if (i < n) p[i] = v;
}

// a_s[n,h] = <h[n,h,:], att_src[h,:]>,  a_d likewise
__global__ void att_scores(const float* __restrict__ Hb,
                           const float* __restrict__ att_s,
                           const float* __restrict__ att_d,
                           float* __restrict__ as_, float* __restrict__ ad_,
                           int Hh) {
  int i = blockIdx.x * blockDim.x + threadIdx.x;
  if (i >= N_NODES * Hh) return;
  const int n = i / Hh, h = i % Hh;
  const float* hp = Hb + ((size_t)n * Hh + h) * DHEAD;
  const float* sp = att_s + h * DHEAD;
  const float* dp = att_d + h * DHEAD;
  float ss = 0.f, sd = 0.f;
  #pragma unroll 4
  for (int d = 0; d < DHEAD; ++d) { float v = hp[d]; ss += v * sp[d]; sd += v * dp[d]; }
  as_[i] = ss; ad_[i] = sd;
}

__global__ void edge_max(const long long* __restrict__ ei,
                         const float* __restrict__ as_, const float* __restrict__ ad_,
                         unsigned* __restrict__ mbits, int Hh) {
  const int E2 = N_EDGES + N_NODES;
  int i = blockIdx.x * blockDim.x + threadIdx.x;
  if (i >= E2 * Hh) return;
  int e = i / Hh, h = i % Hh, s, d;
  edge_sd(ei, e, s, d);
  float v = leaky02(as_[s * Hh + h] + ad_[d * Hh + h]);
  atomicMax(&mbits[d * Hh + h], f2ord(v));
}

__global__ void edge_sum(const long long* __restrict__ ei,
                         const float* __restrict__ as_, const float* __restrict__ ad_,
                         const unsigned* __restrict__ mbits, float* __restrict__ den,
                         int Hh) {
  const int E2 = N_EDGES + N_NODES;
  int i = blockIdx.x * blockDim.x + threadIdx.x;
  if (i >= E2 * Hh) return;
  int e = i / Hh, h = i % Hh, s, d;
  edge_sd(ei, e, s, d);
  float v = leaky02(as_[s * Hh + h] + ad_[d * Hh + h]);
  atomicAdd(&den[d * Hh + h], __expf(v - ord2f(mbits[d * Hh + h])));
}

// one wave per edge; lanes stride the H*128 feature columns
__global__ void edge_scatter(const long long* __restrict__ ei,
                             const float* __restrict__ as_, const float* __restrict__ ad_,
                             const unsigned* __restrict__ mbits, const float* __restrict__ den,
                             const float* __restrict__ Hb, float* __restrict__ outb,
                             int Hh) {
  const int E2 = N_EDGES + N_NODES;
  int w = blockIdx.x * (blockDim.x >> 5) + (threadIdx.x >> 5);
  int lane = threadIdx.x & 31;
  if (w >= E2) return;
  int s, d;
  edge_sd(ei, w, s, d);
  const int HD = Hh * DHEAD;
  for (int col = lane; col < HD; col += 32) {
    const int h = col >> 7;                       // DHEAD == 128
    float v = leaky02(as_[s * Hh + h] + ad_[d * Hh + h]);
    float alpha = __expf(v - ord2f(mbits[d * Hh + h])) / (den[d * Hh + h] + 1e-16f);
    atomicAdd(&outb[(size_t)d * HD + col], Hb[(size_t)s * HD + col] * alpha);
  }
}

__global__ void bias_elu(float* __restrict__ out, const float* __restrict__ bias,
                         size_t total, int HD) {
  size_t i = (size_t)blockIdx.x * blockDim.x + threadIdx.x;
  if (i >= total) return;
  float v = out[i] + bias[i % HD];
  out[i] = v > 0.0f ? v : (__expf(v) - 1.0f);
}

__global__ void pool_sum(const float* __restrict__ h3, const long long* __restrict__ batch,
                         float* __restrict__ pooled, float* __restrict__ cnt) {
  int i = blockIdx.x * blockDim.x + threadIdx.x;
  if (i >= N_NODES * DHEAD) return;
  int n = i >> 7, d = i & 127;
  int g = (int)batch[n];
  atomicAdd(&pooled[g * DHEAD + d], h3[i]);
  if (d == 0) atomicAdd(&cnt[g], 1.0f);
}

__global__ void final_fc(const float* __restrict__ pooled, const float* __restrict__ cnt,
                         const float* __restrict__ fcW, const float* __restrict__ fcb,
                         float* __restrict__ out) {
  int i = threadIdx.x;                 // 64 graphs x 3 classes = 192
  if (i >= N_GRAPHS * 3) return;
  int g = i / 3, j = i % 3;
  float c = fmaxf(cnt[g], 1.0f);
  float acc = fcb[j];
  #pragma unroll 4
  for (int d = 0; d < DHEAD; ++d) acc += (pooled[g * DHEAD + d] / c) * fcW[d * 3 + j];
  out[i] = acc;
}

// ---------------------------------------------------------------------------
static inline int ceil_div(long long a, long long b) { return (int)((a + b - 1) / b); }

static void run_layer(const float* Xl, const float* W, const float* att_s,
                      const float* att_d, const float* bias, int Fin, int Hh,
                      const long long* ei, float* hB, float* outA,
                      float* as_, float* ad_, unsigned* mbits, float* den,
                      hipStream_t stream) {
  const int Fout = Hh * DHEAD;
  dim3 g(N_NODES / 16, Fout / 128);
  gemm_wmma_f32<<<g, 256, 0, stream>>>(Xl, W, hB, Fin, Fout);

  const int nh = N_NODES * Hh;
  att_scores<<<ceil_div(nh, 256), 256, 0, stream>>>(hB, att_s, att_d, as_, ad_, Hh);

  const size_t tot = (size_t)N_NODES * Fout;
  fill_f32<<<ceil_div((long long)tot, 256), 256, 0, stream>>>(outA, 0.0f, tot);
  fill_u32<<<ceil_div(nh, 256), 256, 0, stream>>>(mbits, NEG_INF_ORD, (size_t)nh);
  fill_f32<<<ceil_div(nh, 256), 256, 0, stream>>>(den, 0.0f, (size_t)nh);

  const int E2 = N_EDGES + N_NODES;
  edge_max<<<ceil_div((long long)E2 * Hh, 256), 256, 0, stream>>>(ei, as_, ad_, mbits, Hh);
  edge_sum<<<ceil_div((long long)E2 * Hh, 256), 256, 0, stream>>>(ei, as_, ad_, mbits, den, Hh);
  edge_scatter<<<ceil_div(E2, 8), 256, 0, stream>>>(ei, as_, ad_, mbits, den, hB, outA, Hh);
  bias_elu<<<ceil_div((long long)tot, 256), 256, 0, stream>>>(outA, bias, tot, Fout);
}

extern "C" void kernel_launch(void* const* d_in, const int* in_sizes, int n_in,
                              void* d_out, int out_size, void* d_ws, size_t ws_size,
                              hipStream_t stream) {
  const float*     x     = (const float*)d_in[0];
  const long long* ei    = (const long long*)d_in[1];
  const long long* batch = (const long long*)d_in[2];
  const float* W1  = (const float*)d_in[3];
  const float* as1 = (const float*)d_in[4];
  const float* ad1 = (const float*)d_in[5];
  const float* b1  = (const float*)d_in[6];
  const float* W2  = (const float*)d_in[7];
  const float* as2 = (const float*)d_in[8];
  const float* ad2 = (const float*)d_in[9];
  const float* b2  = (const float*)d_in[10];
  const float* W3  = (const float*)d_in[11];
  const float* as3 = (const float*)d_in[12];
  const float* ad3 = (const float*)d_in[13];
  const float* b3  = (const float*)d_in[14];
  const float* fcW = (const float*)d_in[15];
  const float* fcb = (const float*)d_in[16];

  // workspace layout (floats): two 20000x512 ping-pong buffers + small arrays
  const size_t BUF = (size_t)N_NODES * 512;
  float*    bufA   = (float*)d_ws;              // X / scatter-out / ELU in place
  float*    bufB   = bufA + BUF;                // GEMM output h
  float*    as_    = bufB + BUF;
  float*    ad_    = as_ + (size_t)N_NODES * 4;
  unsigned* mbits  = (unsigned*)(ad_ + (size_t)N_NODES * 4);
  float*    den    = (float*)(mbits + (size_t)N_NODES * 4);
  float*    pooled = den + (size_t)N_NODES * 4;
  float*    cnt    = pooled + N_GRAPHS * DHEAD;

  run_layer(x,    W1, as1, ad1, b1, 384, 4, ei, bufB, bufA, as_, ad_, mbits, den, stream);
  run_layer(bufA, W2, as2, ad2, b2, 512, 2, ei, bufB, bufA, as_, ad_, mbits, den, stream);
  run_layer(bufA, W3, as3, ad3, b3, 256, 1, ei, bufB, bufA, as_, ad_, mbits, den, stream);

  fill_f32<<<ceil_div(N_GRAPHS * DHEAD, 256), 256, 0, stream>>>(pooled, 0.0f, N_GRAPHS * DHEAD);
  fill_f32<<<1, 64, 0, stream>>>(cnt, 0.0f, N_GRAPHS);
  pool_sum<<<ceil_div(N_NODES * DHEAD, 256), 256, 0, stream>>>(bufA, batch, pooled, cnt);
  final_fc<<<1, 192, 0, stream>>>(pooled, cnt, fcW, fcb, (float*)d_out);
}